// BiasingGateB_55679956025637
// MI455X (gfx1250) — compile-verified
//
#include <hip/hip_runtime.h>
#include <hip/hip_bf16.h>
#include <math.h>

typedef __attribute__((ext_vector_type(2))) float v2f;
typedef __attribute__((ext_vector_type(4))) float v4f;
typedef __attribute__((ext_vector_type(8))) float v8f;

#define B_   64
#define T_   2048
#define D_   1024
#define H_   16
#define M_   64
#define TCHUNK 128   // T_ / 16 chunks

// ---------------------------------------------------------------------------
// 0) zero the accumulator (harness poisons d_ws with 0xAA)
// ---------------------------------------------------------------------------
__global__ void bg_zero_kernel(float* __restrict__ p, int n) {
    int i = blockIdx.x * blockDim.x + threadIdx.x;
    if (i < n) p[i] = 0.0f;
}

// ---------------------------------------------------------------------------
// 1) mean-pool accumulate: sum[b][d] += sum_t x[b][t][d]
//    This is the 512 MiB stream -> pure HBM bandwidth. Non-temporal loads,
//    float4-wide, fully coalesced (256 threads * 16B = 4KB = one D row).
//    grid = (B, 16 T-chunks); 16-way atomic contention per sum element.
// ---------------------------------------------------------------------------
__global__ void bg_pool_kernel(const float* __restrict__ x, float* __restrict__ sum) {
    const int b   = blockIdx.x;              // 0..63
    const int tc  = blockIdx.y;              // 0..15
    const int tid = threadIdx.x;             // 0..255 ; covers D via v4f
    const v4f* xb = (const v4f*)(x + (size_t)b * T_ * D_);
    const int t0 = tc * TCHUNK;

    v4f acc = {0.0f, 0.0f, 0.0f, 0.0f};
#pragma unroll 8
    for (int t = 0; t < TCHUNK; ++t) {
        v4f v = __builtin_nontemporal_load(&xb[(size_t)(t0 + t) * (D_ / 4) + tid]);
        acc += v;
    }
    float* s = sum + b * D_ + tid * 4;
    atomicAdd(s + 0, acc.x);
    atomicAdd(s + 1, acc.y);
    atomicAdd(s + 2, acc.z);
    atomicAdd(s + 3, acc.w);
}

// ---------------------------------------------------------------------------
// 2) projection on the matrix pipe: inp = (sum/T) @ W_p + b_p   [64 x 1024]
//    fp32 WMMA 16x16x4, one wave per 16x16 output tile, K-loop of 256 steps.
//    A (16x4 fp32): VGPR0 = K0 (lanes 0-15) / K2 (lanes 16-31); VGPR1 = K1/K3.
//    B (4x16 fp32): mirrored layout, lane = N.
//    C/D (16x16 f32): VGPR r -> M=r (lanes 0-15), M=r+8 (lanes 16-31), N=lane%16.
// ---------------------------------------------------------------------------
__global__ void bg_proj_wmma_kernel(const float* __restrict__ sum,
                                    const float* __restrict__ Wp,
                                    const float* __restrict__ bp,
                                    float* __restrict__ inp) {
    const int tile = blockIdx.x;       // 0..255
    const int tm   = tile >> 6;        // 0..3   (M tiles: 64/16)
    const int tn   = tile & 63;        // 0..63  (N tiles: 1024/16)
    const int lane = threadIdx.x;      // 0..31
    const int half = lane >> 4;        // 0 or 1
    const int l16  = lane & 15;

    const float invT  = 1.0f / (float)T_;
    const int   row_a = tm * 16 + l16;       // M index held by this lane (A)
    const int   col_b = tn * 16 + l16;       // N index held by this lane (B/D)

    v8f c = {};
    for (int k0 = 0; k0 < D_; k0 += 4) {
        const int ka = k0 + half * 2;
        v2f a, bb;
        a.x  = sum[row_a * D_ + ka + 0] * invT;   // K = ka
        a.y  = sum[row_a * D_ + ka + 1] * invT;   // K = ka+1
        bb.x = Wp[(size_t)(ka + 0) * D_ + col_b];
        bb.y = Wp[(size_t)(ka + 1) * D_ + col_b];
        c = __builtin_amdgcn_wmma_f32_16x16x4_f32(
                /*neg_a=*/false, a, /*neg_b=*/false, bb,
                /*c_mod=*/(short)0, c, /*reuse_a=*/false, /*reuse_b=*/false);
    }

#pragma unroll
    for (int r = 0; r < 8; ++r) {
        const int row = tm * 16 + half * 8 + r;
        inp[row * D_ + col_b] = c[r] + bp[col_b];
    }
}

// ---------------------------------------------------------------------------
// 3) cosine sim vs 64 patterns, argmax (first-max tie-break like jnp.argmax),
//    sigmoid gate, gather bias row or zeros. One block per batch row.
// ---------------------------------------------------------------------------
__global__ void bg_gate_kernel(const float* __restrict__ inp,
                               const float* __restrict__ patterns,
                               const float* __restrict__ bias_table,
                               float* __restrict__ out) {
    const int b = blockIdx.x;    // 0..63
    const int m = threadIdx.x;   // 0..63 (one pattern per thread)

    __shared__ float s_sim[M_];
    __shared__ int   s_idx;
    __shared__ float s_score;

    const float* ib = inp + b * D_;
    const float* pm = patterns + m * D_;

    float dot = 0.0f, pn = 0.0f, inn = 0.0f;
    for (int d = 0; d < D_; ++d) {
        const float iv = ib[d];
        const float pv = pm[d];
        dot += iv * pv;
        pn  += pv * pv;
        inn += iv * iv;
    }
    s_sim[m] = dot / ((sqrtf(inn) + 1e-8f) * (sqrtf(pn) + 1e-8f));
    __syncthreads();

    if (m == 0) {
        float best = s_sim[0];
        int   bi   = 0;
        for (int j = 1; j < M_; ++j) {
            if (s_sim[j] > best) { best = s_sim[j]; bi = j; }  // strict > : first max wins
        }
        s_idx = bi;
        s_score = best;
    }
    __syncthreads();

    if (m < H_) {
        const float gate = 1.0f / (1.0f + __expf(-s_score));
        out[b * H_ + m] = (gate > 0.8f) ? bias_table[s_idx * H_ + m] : 0.0f;
    }
}

// ---------------------------------------------------------------------------
// launch
// ---------------------------------------------------------------------------
extern "C" void kernel_launch(void* const* d_in, const int* in_sizes, int n_in,
                              void* d_out, int out_size, void* d_ws, size_t ws_size,
                              hipStream_t stream) {
    const float* x          = (const float*)d_in[0];
    // d_in[1] = xa : unused by the reference
    const float* Wp         = (const float*)d_in[2];
    const float* bp         = (const float*)d_in[3];
    const float* patterns   = (const float*)d_in[4];
    const float* bias_table = (const float*)d_in[5];
    float*       out        = (float*)d_out;

    float* sum = (float*)d_ws;          // 64*1024 floats (256 KB)
    float* inp = sum + B_ * D_;         // 64*1024 floats (256 KB)

    bg_zero_kernel<<<(B_ * D_ + 255) / 256, 256, 0, stream>>>(sum, B_ * D_);

    dim3 pg(B_, T_ / TCHUNK);           // 64 x 16 blocks
    bg_pool_kernel<<<pg, 256, 0, stream>>>(x, sum);

    bg_proj_wmma_kernel<<<(B_ / 16) * (D_ / 16), 32, 0, stream>>>(sum, Wp, bp, inp);

    bg_gate_kernel<<<B_, M_, 0, stream>>>(inp, patterns, bias_table, out);
}